// Net_38500086842054
// MI455X (gfx1250) — compile-verified
//
#include <hip/hip_runtime.h>

// ---------------------------------------------------------------------------
// Nadaraya-Watson kernel regression with learned 4->3 projection.
//   Xw = x @ W^T          [4096,3]   (WMMA f32 16x16x4)
//   Z  = train_X @ W^T    [8192,3]   (WMMA f32 16x16x4)
//   k[n,b,d]   = exp2( (c2/h^2) * (Z[n,d]-Xw[b,d])^2 ),  c2 = -0.5*log2(e)
//   out[b,d]   = sum_n k*Y[n] / sum_n k   (fixed-order 2-pass reduction)
//
// Compute-bound analysis (MI455X): inputs total ~300 KB (nanoseconds at
// 23.3 TB/s); the cost is ~100M v_exp_f32 evaluations. So: stage projected
// train rows in LDS as packed float4 (one ds_load_b128 broadcast/iter),
// keep the loop pure VALU+TRANS, and maximize wave count (2048 waves).
// ---------------------------------------------------------------------------

typedef __attribute__((ext_vector_type(2))) float v2f;
typedef __attribute__((ext_vector_type(8))) float v8f;

#define N_TRAIN   8192
#define B_QUERY   4096
#define D_IN      4
#define D_OUT     3
#define CHUNK     512                     // train rows per block (LDS tile)
#define NCHUNK    (N_TRAIN / CHUNK)       // 16
#define BLOCK_B   256                     // query rows per block (1 per thread)
#define NB_BLOCKS (B_QUERY / BLOCK_B)     // 16
#define OUT_ELEMS (B_QUERY * D_OUT)       // 12288

#if __has_builtin(__builtin_amdgcn_exp2f)
#define EXP2F(x) __builtin_amdgcn_exp2f(x)
#else
#define EXP2F(x) exp2f(x)
#endif

// ---------------------------------------------------------------------------
// Projection: one wave per 16-row tile, D = A(16x4) x B(4x16) via
// v_wmma_f32_16x16x4_f32.  B = W^T zero-padded to 4x16 columns.
//
// A layout (32-bit A 16x4, ISA 7.12.2): lanes 0-15 -> M=lane, VGPR0=K0,
// VGPR1=K1; lanes 16-31 -> M=lane-16, VGPR0=K2, VGPR1=K3.
// B layout (4x16, column per lane): lanes 0-15 -> K=0,1; lanes 16-31 -> K=2,3.
// D layout (16x16 f32): VGPR r -> M = r + 8*(lane>=16), N = lane&15.
// ---------------------------------------------------------------------------
__global__ __launch_bounds__(32)
void proj16x4_wmma(const float* __restrict__ X, const float* __restrict__ W,
                   float* __restrict__ out) {
  const int lane = threadIdx.x;          // 0..31
  const int half = lane >> 4;            // 0 or 1
  const int m    = lane & 15;
  const int row0 = blockIdx.x * 16;

  // A: 16 rows of X, K striped across lane halves
  v2f a;
  a.x = X[(row0 + m) * D_IN + 2 * half + 0];
  a.y = X[(row0 + m) * D_IN + 2 * half + 1];

  // B: column n of W^T (i.e. row n of W), zero for padded columns n>=3.
  // Branch-free so EXEC stays all-1s into the WMMA.
  const int   n   = m;
  const int   nn  = (n < D_OUT) ? n : 0;
  const float msk = (n < D_OUT) ? 1.0f : 0.0f;
  v2f b;
  b.x = msk * W[nn * D_IN + 2 * half + 0];
  b.y = msk * W[nn * D_IN + 2 * half + 1];

  v8f c = {};
  c = __builtin_amdgcn_wmma_f32_16x16x4_f32(
      /*neg_a=*/false, a, /*neg_b=*/false, b,
      /*c_mod=*/(short)0, c, /*reuse_a=*/false, /*reuse_b=*/false);

  if (n < D_OUT) {
#pragma unroll
    for (int r = 0; r < 8; ++r) {
      out[(row0 + r + 8 * half) * D_OUT + n] = c[r];
    }
  }
}

// ---------------------------------------------------------------------------
// Heavy kernel: each block = 256 query rows x 512 train rows.
// Train rows staged in LDS as packed float4 {z0,z1,z2,y} -> one
// ds_load_b128 broadcast per inner iteration (uniform address, no bank
// conflicts). Inner loop: 3x (sub, mul, v_exp_f32, fma, add).
// Partial sums written to fixed slots (bit-deterministic).
// ---------------------------------------------------------------------------
__global__ __launch_bounds__(BLOCK_B)
void nw_partial(const float* __restrict__ Xw, const float* __restrict__ Z,
                const float* __restrict__ Y, const float* __restrict__ hptr,
                float* __restrict__ pnum, float* __restrict__ pden) {
  __shared__ float4 sZY[CHUNK];          // 8 KB

  const int t    = threadIdx.x;
  const int bblk = blockIdx.x;           // 0..15
  const int cblk = blockIdx.y;           // 0..15
  const int n0   = cblk * CHUNK;

  for (int i = t; i < CHUNK; i += BLOCK_B) {
    const int n = n0 + i;
    sZY[i] = make_float4(Z[n * D_OUT + 0], Z[n * D_OUT + 1],
                         Z[n * D_OUT + 2], Y[n]);
  }
  __syncthreads();

  const int   bq    = bblk * BLOCK_B + t;
  const float inv_h = 1.0f / hptr[0];
  // exp(-0.5*u^2) = exp2(c2 * u^2); fold 1/h^2 in: s = c2 / h^2
  const float s = -0.72134752044448170368f * inv_h * inv_h;

  const float xw0 = Xw[bq * D_OUT + 0];
  const float xw1 = Xw[bq * D_OUT + 1];
  const float xw2 = Xw[bq * D_OUT + 2];

  float num0 = 0.f, num1 = 0.f, num2 = 0.f;
  float den0 = 0.f, den1 = 0.f, den2 = 0.f;

#pragma unroll 4
  for (int i = 0; i < CHUNK; ++i) {
    const float4 zy = sZY[i];
    const float u0 = zy.x - xw0;
    const float u1 = zy.y - xw1;
    const float u2 = zy.z - xw2;
    const float e0 = EXP2F(s * u0 * u0);
    const float e1 = EXP2F(s * u1 * u1);
    const float e2 = EXP2F(s * u2 * u2);
    num0 = __builtin_fmaf(e0, zy.w, num0);  den0 += e0;
    num1 = __builtin_fmaf(e1, zy.w, num1);  den1 += e1;
    num2 = __builtin_fmaf(e2, zy.w, num2);  den2 += e2;
  }

  const int base = (cblk * B_QUERY + bq) * D_OUT;
  pnum[base + 0] = num0;  pden[base + 0] = den0;
  pnum[base + 1] = num1;  pden[base + 1] = den1;
  pnum[base + 2] = num2;  pden[base + 2] = den2;
}

// ---------------------------------------------------------------------------
// Finalize: fixed-order sum over the 16 chunks, then divide.
// ---------------------------------------------------------------------------
__global__ __launch_bounds__(256)
void nw_final(const float* __restrict__ pnum, const float* __restrict__ pden,
              float* __restrict__ out) {
  const int idx = blockIdx.x * blockDim.x + threadIdx.x;
  if (idx >= OUT_ELEMS) return;
  float num = 0.f, den = 0.f;
#pragma unroll
  for (int c = 0; c < NCHUNK; ++c) {
    num += pnum[c * OUT_ELEMS + idx];
    den += pden[c * OUT_ELEMS + idx];
  }
  out[idx] = num / den;
}

// ---------------------------------------------------------------------------
// Host launch. Inputs (setup_inputs order): x, train_X, Y, W, h.
// Workspace layout (floats):
//   [0, 12288)                     Xw
//   [12288, 36864)                 Z
//   [36864,  36864+196608)         pnum (16 chunks x 12288)
//   [233472, 233472+196608)        pden
// Total ~1.72 MB.
// ---------------------------------------------------------------------------
extern "C" void kernel_launch(void* const* d_in, const int* in_sizes, int n_in,
                              void* d_out, int out_size, void* d_ws, size_t ws_size,
                              hipStream_t stream) {
  (void)in_sizes; (void)n_in; (void)out_size; (void)ws_size;

  const float* x      = (const float*)d_in[0];
  const float* trainX = (const float*)d_in[1];
  const float* Y      = (const float*)d_in[2];
  const float* W      = (const float*)d_in[3];
  const float* h      = (const float*)d_in[4];
  float*       out    = (float*)d_out;

  float* ws   = (float*)d_ws;
  float* Xw   = ws;                        // 12288 floats
  float* Z    = ws + OUT_ELEMS;            // 24576 floats
  float* pnum = Z + N_TRAIN * D_OUT;       // 196608 floats
  float* pden = pnum + NCHUNK * OUT_ELEMS; // 196608 floats

  // Projections via WMMA: one wave per 16-row tile.
  proj16x4_wmma<<<B_QUERY / 16, 32, 0, stream>>>(x, W, Xw);
  proj16x4_wmma<<<N_TRAIN / 16, 32, 0, stream>>>(trainX, W, Z);

  // Heavy exp/reduce pass: 16 b-blocks x 16 n-chunks = 256 blocks,
  // 256 threads each -> 2048 wave32s.
  nw_partial<<<dim3(NB_BLOCKS, NCHUNK), BLOCK_B, 0, stream>>>(
      Xw, Z, Y, h, pnum, pden);

  // Deterministic finalize.
  nw_final<<<(OUT_ELEMS + 255) / 256, 256, 0, stream>>>(pnum, pden, out);
}